// EigVals_17154099380508
// MI455X (gfx1250) — compile-verified
//
#include <hip/hip_runtime.h>
#include <math.h>

// Eigenvalues of per-voxel symmetric 3x3 matrices.
// Input  X: (2, 9, 96,96,96) fp32  -> 9 contiguous planes of n floats per batch
// Output Y: (2, 3, 96,96,96) fp32  -> 3 contiguous planes (ascending eigenvalues)
//
// Memory-bound streaming kernel (85MB @ 23.3TB/s ~ 3.6us floor).
// CDNA5 path: global_load_async_to_lds_b128 staging (ASYNCcnt), then LDS->VGPR
// b128 reads, closed-form trigonometric eigensolve, b128 stores.
// Grid is 2D (x = tile, y = batch) so there is no scalar integer division in
// the wave preamble and the async copies issue immediately after launch.

#define TPB  256
#define VPT  4
#define TILE (TPB * VPT)   // 1024 voxels per block, 36KB LDS (9 channels)

// Closed-form symmetric 3x3 eigenvalues (Smith's trigonometric method).
// Inputs are the raw (possibly non-symmetric) row-major entries; symmetrized here.
__device__ __forceinline__ void eig3(float m00, float m01, float m02,
                                     float m10, float m11, float m12,
                                     float m20, float m21, float m22,
                                     float& lo, float& mi, float& hi) {
    const float a01 = 0.5f * (m01 + m10);
    const float a02 = 0.5f * (m02 + m20);
    const float a12 = 0.5f * (m12 + m21);

    const float q  = (m00 + m11 + m22) * (1.0f / 3.0f);
    const float d0 = m00 - q, d1 = m11 - q, d2 = m22 - q;
    const float p1 = a01 * a01 + a02 * a02 + a12 * a12;
    const float p2 = d0 * d0 + d1 * d1 + d2 * d2 + 2.0f * p1;

    const float s    = p2 * (1.0f / 6.0f);
    const float rs   = rsqrtf(s);                 // v_rsq_f32
    const bool  ok   = s > 1e-30f;
    const float p    = ok ? s * rs : 0.0f;        // sqrt(p2/6)
    const float pinv = ok ? rs : 0.0f;            // 1/p

    const float b00 = d0 * pinv, b11 = d1 * pinv, b22 = d2 * pinv;
    const float b01 = a01 * pinv, b02 = a02 * pinv, b12 = a12 * pinv;

    float detB = b00 * (b11 * b22 - b12 * b12)
               - b01 * (b01 * b22 - b12 * b02)
               + b02 * (b01 * b12 - b11 * b02);
    float r = 0.5f * detB;
    r = fminf(1.0f, fmaxf(-1.0f, r));

    const float phi  = acosf(r) * (1.0f / 3.0f);          // phi in [0, pi/3]
    const float cph  = __cosf(phi);                        // hardware v_cos_f32
    const float sph  = __builtin_sqrtf(fmaxf(0.0f, 1.0f - cph * cph)); // sin>=0
    // cos(phi + 2pi/3) = -1/2*cos(phi) - sqrt(3)/2*sin(phi)
    const float cmin = -0.5f * cph - 0.86602540378443865f * sph;

    hi = q + 2.0f * p * cph;    // largest
    lo = q + 2.0f * p * cmin;   // smallest
    mi = 3.0f * q - hi - lo;    // middle via trace identity
}

__global__ __launch_bounds__(TPB)
void eig3x3_kernel(const float* __restrict__ X, float* __restrict__ Y, int n) {
    __shared__ float tile[9][TILE];   // 36 KB -> 8 blocks (64 waves) per WGP

    const int t  = (int)threadIdx.x;
    const int tb = (int)blockIdx.x;   // tile index within batch
    const int bi = (int)blockIdx.y;   // batch index
    const int v0 = tb * TILE;

    const size_t inb  = (size_t)bi * 9u * (size_t)n;
    const size_t outb = (size_t)bi * 3u * (size_t)n;

    if (v0 + TILE <= n) {
        // ---- Fast path: async-stage the 9-channel tile into LDS ----
        // Each lane fetches 16B per channel; it later reads back exactly those
        // bytes, so only an ASYNCcnt wait is needed (no workgroup barrier).
        const unsigned lds0 =
            (unsigned)(uintptr_t)(&tile[0][0]) + (unsigned)(t * VPT) * 4u;
        const unsigned g0 =
            (unsigned)((inb + (size_t)v0 + (size_t)(t * VPT)) * 4u);

#pragma unroll
        for (int c = 0; c < 9; ++c) {
            const unsigned la = lds0 + (unsigned)c * (unsigned)(TILE * 4);
            const unsigned ga = g0 + (unsigned)c * (unsigned)n * 4u;
            // GVS form: vdst = per-lane LDS byte addr, vaddr = 32-bit byte
            // offset, saddr = 64-bit base. Tracked by ASYNCcnt.
            asm volatile("global_load_async_to_lds_b128 %0, %1, %2"
                         :
                         : "v"(la), "v"(ga), "s"(X)
                         : "memory");
        }

        // Address math for the stores overlaps the in-flight async copies.
        const size_t o = outb + (size_t)v0 + (size_t)(t * VPT);

#if __has_builtin(__builtin_amdgcn_s_wait_asynccnt)
        __builtin_amdgcn_s_wait_asynccnt(0);
        asm volatile("" ::: "memory");
#else
        asm volatile("s_wait_asynccnt 0" ::: "memory");
#endif

        // LDS -> VGPR, b128 per channel (16B-aligned, conflict-free stride)
        float4 ch[9];
#pragma unroll
        for (int c = 0; c < 9; ++c)
            ch[c] = *reinterpret_cast<const float4*>(&tile[c][t * VPT]);

        float lo[VPT], mi[VPT], hi[VPT];
#pragma unroll
        for (int j = 0; j < VPT; ++j) {
            float m[9];
#pragma unroll
            for (int c = 0; c < 9; ++c)
                m[c] = reinterpret_cast<const float*>(&ch[c])[j];
            eig3(m[0], m[1], m[2], m[3], m[4], m[5], m[6], m[7], m[8],
                 lo[j], mi[j], hi[j]);
        }

        *reinterpret_cast<float4*>(Y + o) =
            make_float4(lo[0], lo[1], lo[2], lo[3]);
        *reinterpret_cast<float4*>(Y + o + (size_t)n) =
            make_float4(mi[0], mi[1], mi[2], mi[3]);
        *reinterpret_cast<float4*>(Y + o + 2u * (size_t)n) =
            make_float4(hi[0], hi[1], hi[2], hi[3]);
    } else {
        // ---- Tail path (n not a multiple of TILE; unused for 96^3) ----
        const int end = (n < v0 + TILE) ? n : (v0 + TILE);
        for (int idx = v0 + t; idx < end; idx += TPB) {
            float m[9];
#pragma unroll
            for (int c = 0; c < 9; ++c)
                m[c] = X[inb + (size_t)c * (size_t)n + (size_t)idx];
            float lo, mi, hi;
            eig3(m[0], m[1], m[2], m[3], m[4], m[5], m[6], m[7], m[8],
                 lo, mi, hi);
            Y[outb + (size_t)idx]                  = lo;
            Y[outb + (size_t)n + (size_t)idx]      = mi;
            Y[outb + 2u * (size_t)n + (size_t)idx] = hi;
        }
    }
}

extern "C" void kernel_launch(void* const* d_in, const int* in_sizes, int n_in,
                              void* d_out, int out_size, void* d_ws, size_t ws_size,
                              hipStream_t stream) {
    const float* X = (const float*)d_in[0];
    float*       Y = (float*)d_out;

    const int B = 2;                       // matches reference setup (b=2)
    const int n = in_sizes[0] / (B * 9);   // voxels per batch (96^3)

    const int blocks_per_batch = (n + TILE - 1) / TILE;
    dim3 grid((unsigned)blocks_per_batch, (unsigned)B);
    dim3 block(TPB);
    hipLaunchKernelGGL(eig3x3_kernel, grid, block, 0, stream, X, Y, n);
}